// TransEScore_12240656794087
// MI455X (gfx1250) — compile-verified
//
#include <hip/hip_runtime.h>
#include <cstdint>

// TransE edge scoring + segment-sum on MI455X (gfx1250).
// Memory-bound: ~410 MB compulsory traffic vs 0.4 GFLOP -> optimize the
// data path. Wave-per-edge, b128 per lane, async-to-LDS streaming of the
// edge_emb stream (non-temporal so it doesn't evict the L2-resident
// node_emb/h tables), hw f32 atomics for the scatter-add.

#define WAVES_PER_BLOCK 8
#define EDGES_PER_WAVE  8
#define EDGES_PER_BLOCK (WAVES_PER_BLOCK * EDGES_PER_WAVE)
#define DIM 128
#define GAMMA_F 12.0f

// GLOBAL_LOAD_ASYNC_TO_LDS_B128, GVS mode:
//   global address is the 64-bit wave-uniform saddr plus the 32-bit per-lane vaddr plus ioffset
//   LDS address is the value in the vdst VGPR (byte offset into wave's LDS allocation)
// Tracked by ASYNCcnt. Non-temporal: edge_emb is a one-shot 307MB stream.
__device__ __forceinline__ void async_row_load_b128_nt(uint32_t lds_byte_off,
                                                       const float* row_base,
                                                       uint32_t lane_byte_off) {
  asm volatile("global_load_async_to_lds_b128 %0, %1, %2 th:TH_LOAD_NT"
               :
               : "v"(lds_byte_off), "v"(lane_byte_off), "s"(row_base)
               : "memory");
}

__device__ __forceinline__ void wait_async0() {
  asm volatile("s_wait_asynccnt 0" ::: "memory");
}

__global__ void __launch_bounds__(256)
transe_main_kernel(const float* __restrict__ node_emb,
                   const float* __restrict__ edge_emb,
                   const int*   __restrict__ src,
                   const int*   __restrict__ dst,
                   float*       __restrict__ h,
                   int E) {
  // 64 edge rows * 512 B = 32 KB LDS; each wave owns an 8-row slice, so a
  // per-wave s_wait_asynccnt suffices (no block barrier).
  __shared__ float4 stage[EDGES_PER_BLOCK][32];

  const int lane  = threadIdx.x & 31;
  const int wave  = threadIdx.x >> 5;
  const int ebase = blockIdx.x * EDGES_PER_BLOCK + wave * EDGES_PER_WAVE;

  // Low 32 bits of a generic pointer to a __shared__ object are the LDS
  // byte offset (flat LDS aperture keeps the offset in addr[31:0]).
  const uint32_t lds0     = (uint32_t)(size_t)(&stage[0][0]);
  const uint32_t lane_off = (uint32_t)lane * 16u;

  // Stage this wave's 8 edge_emb rows into LDS asynchronously.
  #pragma unroll
  for (int i = 0; i < EDGES_PER_WAVE; ++i) {
    int e = ebase + i;
    if (e < E) {
      uint32_t ldsa = lds0 +
          (uint32_t)(wave * EDGES_PER_WAVE + i) * (32u * 16u) + lane_off;
      async_row_load_b128_nt(ldsa, edge_emb + (size_t)e * DIM, lane_off);
    }
  }
  wait_async0();

  #pragma unroll 1
  for (int i = 0; i < EDGES_PER_WAVE; ++i) {
    int e = ebase + i;
    if (e >= E) break;
    int s = src[e];
    int d = dst[e];

    // lane l owns dims [4l, 4l+3]: one b128 per lane = coalesced 512B row.
    const float4 hd = *(const float4*)(node_emb + (size_t)s * DIM + lane * 4);
    const float4 tl = *(const float4*)(node_emb + (size_t)d * DIM + lane * 4);
    const float4 rl = stage[wave * EDGES_PER_WAVE + i][lane];

    float4 tr;
    tr.x = hd.x + rl.x; tr.y = hd.y + rl.y;
    tr.z = hd.z + rl.z; tr.w = hd.w + rl.w;

    float dx = tr.x - tl.x, dy = tr.y - tl.y;
    float dz = tr.z - tl.z, dw = tr.w - tl.w;
    float part = dx * dx + dy * dy + dz * dz + dw * dw;

    // wave32 butterfly reduction -> every lane holds ||diff||^2
    #pragma unroll
    for (int off = 16; off > 0; off >>= 1)
      part += __shfl_xor(part, off, 32);

    float dist  = sqrtf(part);
    float score = 1.0f / (1.0f + __expf(dist - GAMMA_F));  // sigmoid(12-dist)

    float* outp = h + (size_t)d * DIM + lane * 4;
    unsafeAtomicAdd(outp + 0, score * tr.x);
    unsafeAtomicAdd(outp + 1, score * tr.y);
    unsafeAtomicAdd(outp + 2, score * tr.z);
    unsafeAtomicAdd(outp + 3, score * tr.w);
  }
}

__global__ void __launch_bounds__(256)
transe_zero_kernel(float4* __restrict__ p, int n4) {
  int i = blockIdx.x * blockDim.x + threadIdx.x;
  if (i < n4) p[i] = make_float4(0.f, 0.f, 0.f, 0.f);
}

extern "C" void kernel_launch(void* const* d_in, const int* in_sizes, int n_in,
                              void* d_out, int out_size, void* d_ws, size_t ws_size,
                              hipStream_t stream) {
  (void)n_in; (void)d_ws; (void)ws_size;
  const float* node_emb = (const float*)d_in[0];
  const float* edge_emb = (const float*)d_in[1];
  const int*   src      = (const int*)d_in[2];
  const int*   dst      = (const int*)d_in[3];
  float*       out      = (float*)d_out;

  const int E  = in_sizes[2];        // 600000
  const int n4 = out_size / 4;       // N*D/4 float4s

  transe_zero_kernel<<<(n4 + 255) / 256, 256, 0, stream>>>((float4*)out, n4);

  const int blocks = (E + EDGES_PER_BLOCK - 1) / EDGES_PER_BLOCK;  // 9375
  transe_main_kernel<<<blocks, 256, 0, stream>>>(node_emb, edge_emb, src, dst,
                                                 out, E);
}